// Gcn_33337536151790
// MI455X (gfx1250) — compile-verified
//
#include <hip/hip_runtime.h>
#include <math.h>

// ---------------------------------------------------------------------------
// 2-layer GCN for MI455X (gfx1250, wave32, WMMA).
// GEMMs use V_WMMA_F32_16X16X4_F32 (full fp32; problem is HBM/L2-atomic bound,
// so lossy bf16 buys nothing). One wave computes a full 16-row x N output
// strip so the A matrix (X: 205MB > 192MB L2) is streamed from HBM exactly
// once. B strides are compile-time -> immediate-offset loads, no per-step
// address math.
// ---------------------------------------------------------------------------

#define NN     100000      // nodes  (== 6250 * 16, exact M-tile fit)
#define NF     512         // input features
#define NH     64          // hidden
#define NC     40          // classes
#define NCP    48          // padded class count (3 x 16 WMMA tiles)
#define NE     3200000     // edges

typedef __attribute__((ext_vector_type(2))) float v2f;
typedef __attribute__((ext_vector_type(8))) float v8f;

// ------------------------------- degree / norm -----------------------------

__global__ void k_init_deg(float* __restrict__ deg) {
    int i = blockIdx.x * blockDim.x + threadIdx.x;
    if (i < NN) deg[i] = 1.0f;                 // self-loop contribution
}

__global__ void k_deg_edges(const int* __restrict__ dst, float* __restrict__ deg) {
    int e = blockIdx.x * blockDim.x + threadIdx.x;
    if (e < NE) atomicAdd(&deg[dst[e]], 1.0f);
}

__global__ void k_rsqrt(float* __restrict__ d) {
    int i = blockIdx.x * blockDim.x + threadIdx.x;
    if (i < NN) d[i] = rsqrtf(d[i]);           // deg >= 1 always (self-loops)
}

__global__ void k_edge_norm(const int* __restrict__ src, const int* __restrict__ dst,
                            const float* __restrict__ dinv, float* __restrict__ norm) {
    int e = blockIdx.x * blockDim.x + threadIdx.x;
    if (e < NE) norm[e] = dinv[src[e]] * dinv[dst[e]];
}

// zero-pad W2 [NH x NC] -> W2p [NH x NCP]
__global__ void k_pad_w2(const float* __restrict__ W2, float* __restrict__ W2p) {
    int idx = blockIdx.x * blockDim.x + threadIdx.x;
    if (idx >= NH * NCP) return;
    int r = idx / NCP, c = idx - r * NCP;
    W2p[idx] = (c < NC) ? W2[r * NC + c] : 0.0f;
}

// ------------------------------- WMMA GEMM ---------------------------------
// C[M x N] = A[M x K] @ B[K x NB], row-major fp32; NB = N padded to 16*NTN.
// One wave32 per 16-row strip: NTN accumulators, 4 x NTN FLOP-reuse per A
// fragment. K stepped by 4 via v_wmma_f32_16x16x4_f32.
//
// ISA fragment layouts (CDNA5 ISA):
//   A 16x4 : lanes 0-15 -> rows M=lane, VGPR0=K0, VGPR1=K1;
//            lanes 16-31 -> rows M=lane-16, VGPR0=K2, VGPR1=K3.
//   B 4x16 : mirrored (lanes = N columns, halves split K 0..1 / 2..3).
//   C/D    : VGPR r -> row (r + 8*half), col = lane & 15.

template <int K, int N, int NB, int NTN>
__global__ void k_gemm_wmma(const float* __restrict__ A, const float* __restrict__ B,
                            float* __restrict__ C) {
    const int lane = threadIdx.x;              // 0..31 (blockDim.x == 32)
    const int mt   = blockIdx.x * blockDim.y + threadIdx.y;
    if (mt >= NN / 16) return;
    const int m0   = mt << 4;

    const int half = lane >> 4;                // 0: K{0,1}, 1: K{2,3}
    const int l    = lane & 15;

    const float2* ap = reinterpret_cast<const float2*>(
                           A + (size_t)(m0 + l) * K + 2 * half);
    const float*  bp = B + (size_t)(2 * half) * NB + l;   // lane's base column

    v8f acc[NTN];
    #pragma unroll
    for (int t = 0; t < NTN; ++t) acc[t] = {};

    #pragma unroll 4
    for (int k = 0; k < K; k += 4) {
        const float2 av = *ap;  ap += 2;       // A[m][k + {0,1} + 2*half]
        v2f a; a.x = av.x; a.y = av.y;
        #pragma unroll
        for (int t = 0; t < NTN; ++t) {        // compile-time immediate offsets
            v2f b;
            b.x = bp[16 * t];                  // B[k+2h  ][16t + l]
            b.y = bp[16 * t + NB];             // B[k+2h+1][16t + l]
            acc[t] = __builtin_amdgcn_wmma_f32_16x16x4_f32(
                         false, a, false, b, (short)0, acc[t], false, false);
        }
        bp += 4 * NB;
    }

    #pragma unroll
    for (int t = 0; t < NTN; ++t) {
        const int bn = 16 * t + l;
        if (bn < N) {                          // only N=40 tile 2 masks
            float* cp = C + (size_t)(m0 + 8 * half) * N + bn;
            #pragma unroll
            for (int r = 0; r < 8; ++r)
                cp[(size_t)r * N] = acc[t][r];
        }
    }
}

// --------------------------- aggregation (scatter) -------------------------

// agg[i,f] = dinv[i]^2 * h[i,f]   (self-loop term; also initializes agg)
// float4-vectorized: rows are 256B (F=64) / 160B (F=40) -> 16B aligned.
template <int F>
__global__ void k_self_init(const float* __restrict__ h, const float* __restrict__ dinv,
                            float* __restrict__ agg) {
    int idx = blockIdx.x * blockDim.x + threadIdx.x;      // over NN * F/4
    if (idx >= NN * (F / 4)) return;
    int i = idx / (F / 4);
    float d = dinv[i];
    float w = d * d;
    float4 hv = *reinterpret_cast<const float4*>(h + (size_t)idx * 4);
    float4 o;
    o.x = w * hv.x; o.y = w * hv.y; o.z = w * hv.z; o.w = w * hv.w;
    *reinterpret_cast<float4*>(agg + (size_t)idx * 4) = o;
}

// agg[dst, 4q..4q+3] += norm[e] * h[src, 4q..4q+3]
template <int F>
__global__ void k_scatter(const int* __restrict__ src, const int* __restrict__ dst,
                          const float* __restrict__ norm, const float* __restrict__ h,
                          float* __restrict__ agg) {
    const unsigned Q = F / 4;
    unsigned idx = blockIdx.x * blockDim.x + threadIdx.x; // over NE * Q
    if (idx >= (unsigned)NE * Q) return;
    unsigned e = idx / Q;
    unsigned q = idx - e * Q;
    int s = src[e], d = dst[e];
    float w = norm[e];
    float4 hv = *reinterpret_cast<const float4*>(h + (size_t)s * F + 4 * q);
    float* ap = agg + (size_t)d * F + 4 * q;
    atomicAdd(ap + 0, w * hv.x);
    atomicAdd(ap + 1, w * hv.y);
    atomicAdd(ap + 2, w * hv.z);
    atomicAdd(ap + 3, w * hv.w);
}

// --------------------------- bias/relu & softmax ---------------------------

__global__ void k_bias_relu(const float* __restrict__ agg, const float* __restrict__ b,
                            float* __restrict__ out) {
    int idx = blockIdx.x * blockDim.x + threadIdx.x;      // over NN * NH/4
    if (idx >= NN * (NH / 4)) return;
    int f4 = (idx & (NH / 4 - 1)) * 4;
    float4 v = *reinterpret_cast<const float4*>(agg + (size_t)idx * 4);
    float4 o;
    o.x = fmaxf(v.x + b[f4 + 0], 0.0f);
    o.y = fmaxf(v.y + b[f4 + 1], 0.0f);
    o.z = fmaxf(v.z + b[f4 + 2], 0.0f);
    o.w = fmaxf(v.w + b[f4 + 3], 0.0f);
    *reinterpret_cast<float4*>(out + (size_t)idx * 4) = o;
}

__global__ void k_softmax(const float* __restrict__ agg, const float* __restrict__ b2,
                          float* __restrict__ out) {
    int i = blockIdx.x * blockDim.x + threadIdx.x;
    if (i >= NN) return;
    float v[NC];
    float mx = -3.402823466e+38f;
    #pragma unroll
    for (int f = 0; f < NC; ++f) {
        v[f] = agg[(size_t)i * NC + f] + b2[f];
        mx = fmaxf(mx, v[f]);
    }
    float s = 0.0f;
    #pragma unroll
    for (int f = 0; f < NC; ++f) {
        v[f] = __expf(v[f] - mx);
        s += v[f];
    }
    float inv = 1.0f / s;
    #pragma unroll
    for (int f = 0; f < NC; ++f)
        out[(size_t)i * NC + f] = v[f] * inv;
}

// ------------------------------- launcher ----------------------------------

extern "C" void kernel_launch(void* const* d_in, const int* in_sizes, int n_in,
                              void* d_out, int out_size, void* d_ws, size_t ws_size,
                              hipStream_t stream) {
    const float* x  = (const float*)d_in[0];     // [NN, NF]
    const int*   ei = (const int*)d_in[1];       // [2, NE]
    const float* W1 = (const float*)d_in[2];     // [NF, NH]
    const float* b1 = (const float*)d_in[3];     // [NH]
    const float* W2 = (const float*)d_in[4];     // [NH, NC]
    const float* b2 = (const float*)d_in[5];     // [NC]
    float*       out = (float*)d_out;            // [NN, NC]

    const int* src = ei;
    const int* dst = ei + NE;

    // workspace layout (floats)
    float* ws   = (float*)d_ws;
    float* dinv = ws;                            // NN      (padded region)
    float* W2p  = dinv + 100352;                 // NH*NCP  (zero-padded W2)
    float* norm = W2p + 4096;                    // NE
    float* h1   = norm + NE;                     // NN*NH  (reused for relu out)
    float* agg1 = h1 + (size_t)NN * NH;          // NN*NH  (reused for agg2)
    float* h2   = agg1 + (size_t)NN * NH;        // NN*NC
    float* agg2 = agg1;                          // reuse (NN*NC <= NN*NH)

    const int T = 256;
    dim3 gblk(32, 8);                            // 8 wave32 row-strips / block
    const int mtiles = NN / 16;                  // 6250
    const int gemmGrid = (mtiles + 7) / 8;       // 782

    // 1) symmetric normalization (+ pad W2 while we're at it)
    k_init_deg <<<(NN + T - 1) / T, T, 0, stream>>>(dinv);
    k_deg_edges<<<(NE + T - 1) / T, T, 0, stream>>>(dst, dinv);
    k_rsqrt    <<<(NN + T - 1) / T, T, 0, stream>>>(dinv);
    k_edge_norm<<<(NE + T - 1) / T, T, 0, stream>>>(src, dst, dinv, norm);
    k_pad_w2   <<<(NH * NCP + T - 1) / T, T, 0, stream>>>(W2, W2p);

    // 2) h1 = x @ W1 : one HBM pass over X, 4 WMMA tiles per wave
    k_gemm_wmma<NF, NH, NH, 4><<<gemmGrid, gblk, 0, stream>>>(x, W1, h1);

    // 3) agg1 = self-loop + scatter(norm * h1)
    k_self_init<NH><<<(NN * (NH / 4) + T - 1) / T, T, 0, stream>>>(h1, dinv, agg1);
    k_scatter<NH><<<(((unsigned)NE * (NH / 4)) + T - 1) / T, T, 0, stream>>>(src, dst, norm, h1, agg1);

    // 4) h1 <- relu(agg1 + b1)
    k_bias_relu<<<(NN * (NH / 4) + T - 1) / T, T, 0, stream>>>(agg1, b1, h1);

    // 5) h2 = h1 @ W2p : 3 WMMA tiles per wave, stores masked to 40 cols
    k_gemm_wmma<NH, NC, NCP, 3><<<gemmGrid, gblk, 0, stream>>>(h1, W2p, h2);

    // 6) agg2 = self-loop + scatter(norm * h2)
    k_self_init<NC><<<(NN * (NC / 4) + T - 1) / T, T, 0, stream>>>(h2, dinv, agg2);
    k_scatter<NC><<<(((unsigned)NE * (NC / 4)) + T - 1) / T, T, 0, stream>>>(src, dst, norm, h2, agg2);

    // 7) out = softmax(agg2 + b2)
    k_softmax<<<(NN + T - 1) / T, T, 0, stream>>>(agg2, b2, out);
}